// DifferentiableTransformer_44787918962860
// MI455X (gfx1250) — compile-verified
//
#include <hip/hip_runtime.h>
#include <hip/hip_bf16.h>

#define GDIM   96
#define GRID_N (GDIM * GDIM * GDIM)
#define BLOCK  256
#define TBL_MAX 512   // R = 301 in practice; clamp staging to LDS capacity

// gfx1250 async global->LDS copy path (CDNA5-specific), with fallback.
#if defined(__AMDGCN__) && __has_builtin(__builtin_amdgcn_global_load_async_to_lds_b32)
#define USE_ASYNC_LDS 1
typedef __attribute__((address_space(1))) int* as1_i32p;
typedef __attribute__((address_space(3))) int* as3_i32p;
#endif

__global__ __launch_bounds__(BLOCK) void zero_out_kernel(float* __restrict__ out, int n) {
    int i = blockIdx.x * BLOCK + threadIdx.x;
    if (i < n) out[i] = 0.0f;
}

__global__ __launch_bounds__(BLOCK) void dens_scatter_kernel(
    const float* __restrict__ coords,          // (B, A, 3)
    const float* __restrict__ occ,             // (B, A)
    const float* __restrict__ rad,             // (B, A, R)
    const float* __restrict__ g2c,             // (3, 3)
    const unsigned char* __restrict__ act,     // (B, A) bool
    const int* __restrict__ max_extent_p,      // scalar
    float* __restrict__ out,                   // (B, 96, 96, 96)
    int A, int R)
{
    __shared__ float tbl[TBL_MAX];

    const int ba  = blockIdx.x;       // flattened (b, a)
    const int b   = ba / A;
    const int tid = threadIdx.x;
    const int Rc  = (R < TBL_MAX) ? R : TBL_MAX;

    // ---- stage this atom's radial density table into LDS (async DMA path) ----
    const float* __restrict__ rsrc = rad + (size_t)ba * (size_t)R;
#ifdef USE_ASYNC_LDS
    for (int i = tid; i < Rc; i += BLOCK) {
        __builtin_amdgcn_global_load_async_to_lds_b32(
            (as1_i32p)(rsrc + i),
            (as3_i32p)(&tbl[i]),
            0, 0);
    }
#if __has_builtin(__builtin_amdgcn_s_wait_asynccnt)
    __builtin_amdgcn_s_wait_asynccnt(0);
#else
    asm volatile("s_wait_asynccnt 0" ::: "memory");
#endif
#else
    for (int i = tid; i < Rc; i += BLOCK) tbl[i] = rsrc[i];
#endif

    // ---- per-atom uniform data (scalar-load friendly) ----
    const float cx = coords[(size_t)ba * 3 + 0];
    const float cy = coords[(size_t)ba * 3 + 1];
    const float cz = coords[(size_t)ba * 3 + 2];
    const float occv = occ[ba] * (act[ba] ? 1.0f : 0.0f);

    const float m00 = g2c[0], m01 = g2c[1], m02 = g2c[2];
    const float m10 = g2c[3], m11 = g2c[4], m12 = g2c[5];
    const float m20 = g2c[6], m21 = g2c[7], m22 = g2c[8];

    const int      e     = max_extent_p[0];
    const unsigned side  = 2u * (unsigned)e + 1u;
    const unsigned nb    = side * side * side;
    // Exact magic division: for dividend, divisor < 2^16,
    // q = mulhi32(n, 0xFFFFFFFF/side + 1) == n / side.
    const unsigned mside = 0xFFFFFFFFu / side + 1u;

    const float fx = floorf(cx), fy = floorf(cy), fz = floorf(cz);
    // frac + e : delta_x = (frx + e) - ixn, since offset ox = ixn - e
    const float fex = (cx - fx) + (float)e;
    const float fey = (cy - fy) + (float)e;
    const float fez = (cz - fz) + (float)e;
    // floor(coord) - e : base grid corner of the neighbor cube
    const int bx = (int)fx - e, by = (int)fy - e, bz = (int)fz - e;

    __syncthreads();

    if (occv == 0.0f) return;   // uniform across block (after barrier)

    const int maxR = R - 1;
    for (unsigned n = tid; n < nb; n += BLOCK) {
        // n = ((izn*side) + iyn)*side + ixn  (meshgrid 'ij': z, y, x)
        unsigned t   = (unsigned)(((unsigned long long)n * mside) >> 32);
        unsigned ixn = n - t * side;
        unsigned izn = (unsigned)(((unsigned long long)t * mside) >> 32);
        unsigned iyn = t - izn * side;

        // delta = frac - (idx - e)
        float dx = fex - (float)ixn;
        float dy = fey - (float)iyn;
        float dz = fez - (float)izn;

        // cart = grid_to_cartesian * delta
        float c0 = m00 * dx + m01 * dy + m02 * dz;
        float c1 = m10 * dx + m11 * dy + m12 * dz;
        float c2 = m20 * dx + m21 * dy + m22 * dz;
        float dist = sqrtf(c0 * c0 + c1 * c1 + c2 * c2);

        // linear interpolation of radial table at dist / RSTEP
        float r  = dist / 0.01f;
        float fl = floorf(r);
        float wh = r - fl;
        float wl = 1.0f - wh;
        int il = (int)fl;
        il = il < 0 ? 0 : (il > maxR ? maxR : il);
        int ih = il + 1;
        ih = ih > maxR ? maxR : ih;
        float d = wl * tbl[il] + wh * tbl[ih];   // LDS gathers
        float dens = d * occv;

        // wrapped grid point: base + idx, wrapped into [0, 96)
        int gx = bx + (int)ixn; gx += (gx < 0) ? GDIM : 0; gx -= (gx >= GDIM) ? GDIM : 0;
        int gy = by + (int)iyn; gy += (gy < 0) ? GDIM : 0; gy -= (gy >= GDIM) ? GDIM : 0;
        int gz = bz + (int)izn; gz += (gz < 0) ? GDIM : 0; gz -= (gz >= GDIM) ? GDIM : 0;

        size_t flat = (size_t)b * GRID_N
                    + (size_t)gz * (GDIM * GDIM)
                    + (size_t)gy * GDIM
                    + (size_t)gx;

        // native fp32 atomic to L2 (no CAS loop)
        unsafeAtomicAdd(&out[flat], dens);
    }
}

extern "C" void kernel_launch(void* const* d_in, const int* in_sizes, int n_in,
                              void* d_out, int out_size, void* d_ws, size_t ws_size,
                              hipStream_t stream) {
    const float*         coords = (const float*)d_in[0];
    const float*         occ    = (const float*)d_in[1];
    const float*         rad    = (const float*)d_in[2];
    const float*         g2c    = (const float*)d_in[3];
    const unsigned char* act    = (const unsigned char*)d_in[4];
    const int*           me     = (const int*)d_in[5];
    float*               out    = (float*)d_out;

    const int BA = in_sizes[1];                 // B*A
    int B = out_size / GRID_N; if (B < 1) B = 1;
    const int A = BA / B;
    const int R = in_sizes[2] / BA;

    // d_out is poisoned by the harness; zero it every call.
    zero_out_kernel<<<(out_size + BLOCK - 1) / BLOCK, BLOCK, 0, stream>>>(out, out_size);

    // one block per (batch, atom)
    dens_scatter_kernel<<<BA, BLOCK, 0, stream>>>(coords, occ, rad, g2c, act, me,
                                                  out, A, R);
}